// EdgeConv_7129645711688
// MI455X (gfx1250) — compile-verified
//
#include <hip/hip_runtime.h>

// EdgeConv (DGCNN) fused pipeline for MI455X / gfx1250, wave32.
// B=8, C=64, N=4096, O=64, k=20, fp32 throughout (top-k needs fp32 ranking).
//
// Algebraic refactor: y[b,o,n,kk] = W1.x[idx] + (W2-W1).x[n]
//   -> v = W1^T x, u = (W2-W1)^T x  (WMMA f32 16x16x4)
//   -> Gram G = X^T X for kNN        (WMMA f32 16x16x4)
//   -> BN+leaky monotone per channel => max over k from {zmax, zmin}.

typedef float v2f __attribute__((ext_vector_type(2)));
typedef float v8f __attribute__((ext_vector_type(8)));

#define BB 8
#define CC 64
#define NN 4096
#define OO 64
#define KNN 20
#define BN_EPSF 1e-5f
#define SLOPEF 0.2f

__device__ __forceinline__ v8f wmma_f32_4(v2f a, v2f b, v8f c) {
  // D = A(16x4 f32) * B(4x16 f32) + C(16x16 f32)
  return __builtin_amdgcn_wmma_f32_16x16x4_f32(
      /*neg_a=*/false, a, /*neg_b=*/false, b,
      /*c_mod=*/(short)0, c, /*reuse_a=*/false, /*reuse_b=*/false);
}

// ---------------------------------------------------------------- K0: zero BN accumulators
__global__ void k_zero(float* __restrict__ STAT) {
  STAT[threadIdx.x] = 0.0f;  // 128 threads: sum[64] | sumsq[64]
}

// ---------------------------------------------------------------- K1: xx[b,n] = sum_c x^2
__global__ void k_xx(const float* __restrict__ X, float* __restrict__ XX) {
  int pos = blockIdx.x * 256 + threadIdx.x;   // pos = b*N + n
  int b = pos >> 12, n = pos & (NN - 1);
  const float* Xb = X + (size_t)b * CC * NN + n;
  float s = 0.0f;
#pragma unroll
  for (int c = 0; c < CC; ++c) {
    float v = Xb[(size_t)c * NN];
    s += v * v;
  }
  XX[pos] = s;
}

// ---------------------------------------------------------------- K2: u,v projections (WMMA)
// v[b,n,o] = sum_c W[o][c]      * x[b,c,n]
// u[b,n,o] = sum_c (W[o][64+c]-W[o][c]) * x[b,c,n]
// grid = B*64 blocks of 128 threads (4 waves); wave w handles n-tile of 16, all 64 o.
__global__ void k_proj(const float* __restrict__ X, const float* __restrict__ W,
                       float* __restrict__ U, float* __restrict__ V) {
  __shared__ float Wt[64][64];  // Wt[c][o] = W[o][c]
  __shared__ float Wd[64][64];  // Wd[c][o] = W[o][64+c] - W[o][c]
  const int tid = threadIdx.x;
  for (int i = tid; i < 4096; i += 128) {
    int o = i >> 6, c = i & 63;
    float w1 = W[o * 128 + c];
    float w2 = W[o * 128 + 64 + c];
    Wt[c][o] = w1;
    Wd[c][o] = w2 - w1;
  }
  __syncthreads();

  const int b   = blockIdx.x >> 6;
  const int g   = blockIdx.x & 63;
  const int w   = tid >> 5;
  const int lane = tid & 31, l16 = lane & 15, hi = lane >> 4;
  const int n0 = g * 64 + w * 16;
  const float* Xb = X + (size_t)b * CC * NN;

  v8f accV[4] = {};
  v8f accU[4] = {};
#pragma unroll 4
  for (int c0 = 0; c0 < 64; c0 += 4) {
    const int cA = c0 + 2 * hi;  // A 16x4 f32 layout: K=0,1 lanes 0-15; K=2,3 lanes 16-31
    v2f a;
    a.x = Xb[(size_t)cA * NN + n0 + l16];
    a.y = Xb[(size_t)(cA + 1) * NN + n0 + l16];
#pragma unroll
    for (int t = 0; t < 4; ++t) {
      const int o0 = t * 16;
      v2f b1; b1.x = Wt[cA][o0 + l16]; b1.y = Wt[cA + 1][o0 + l16];
      accV[t] = wmma_f32_4(a, b1, accV[t]);
      v2f b2; b2.x = Wd[cA][o0 + l16]; b2.y = Wd[cA + 1][o0 + l16];
      accU[t] = wmma_f32_4(a, b2, accU[t]);
    }
  }
  // C/D layout: lane<16 -> (M=r, Nc=lane); lane>=16 -> (M=r+8, Nc=lane-16)
#pragma unroll
  for (int t = 0; t < 4; ++t) {
    const int o0 = t * 16;
#pragma unroll
    for (int r = 0; r < 8; ++r) {
      size_t row = (size_t)b * NN + n0 + r + 8 * hi;
      V[row * 64 + o0 + l16] = accV[t][r];
      U[row * 64 + o0 + l16] = accU[t][r];
    }
  }
}

// ---------------------------------------------------------------- K3: Gram (WMMA) + fused top-k
// One WG = 16 query rows x one batch. 8 waves compute 16x512 Gram chunks into LDS;
// 256 threads (16 per row) stream-insert into register-resident sorted top-20 lists;
// final per-row 16-way merge. Rank key: s = <xi,xj> - 0.5*xx_j (monotone in pdist).
#define GSTR 516  // padded LDS row stride (floats) to split banks between half-waves
__global__ void k_knn(const float* __restrict__ X, const float* __restrict__ XX,
                      int* __restrict__ IDX) {
  __shared__ __align__(16) char smem[16 * 16 * KNN * 8];  // 40960B >= Gbuf 33024B
  float* Gbuf = (float*)smem;                       // [16][GSTR] during chunk phases
  float* Mv   = (float*)smem;                       // [16][16][KNN] merge values
  int*   Mi   = (int*)(smem + 16 * 16 * KNN * 4);   // [16][16][KNN] merge indices

  const int tid = threadIdx.x;
  const int w = tid >> 5, lane = tid & 31, l16 = lane & 15, hi = lane >> 4;
  const int b  = blockIdx.y;
  const int i0 = blockIdx.x * 16;
  const float* Xb  = X  + (size_t)b * CC * NN;
  const float* XXb = XX + (size_t)b * NN;

  // A matrix (rows i0..i0+15, full K=64) resident in registers: 16 x v2f
  v2f aK[16];
#pragma unroll
  for (int kk = 0; kk < 16; ++kk) {
    const int cA = 4 * kk + 2 * hi;
    aK[kk].x = Xb[(size_t)cA * NN + i0 + l16];
    aK[kk].y = Xb[(size_t)(cA + 1) * NN + i0 + l16];
  }

  float bv[KNN];
  int   bi[KNN];
#pragma unroll
  for (int m = 0; m < KNN; ++m) { bv[m] = -3.4e38f; bi[m] = -1; }

  const int row = tid >> 4;  // selection mapping: 16 threads per row
  const int sub = tid & 15;

  for (int ch = 0; ch < 8; ++ch) {
    // --- compute phase: wave w fills columns [w*64, w*64+64) of this 512-col chunk
#pragma unroll
    for (int t = 0; t < 4; ++t) {
      const int jt = ch * 512 + w * 64 + t * 16;
      v8f acc = {};
#pragma unroll
      for (int kk = 0; kk < 16; ++kk) {
        const int cA = 4 * kk + 2 * hi;
        v2f bbm;
        bbm.x = Xb[(size_t)cA * NN + jt + l16];
        bbm.y = Xb[(size_t)(cA + 1) * NN + jt + l16];
        acc = wmma_f32_4(aK[kk], bbm, acc);
      }
      const int colb = w * 64 + t * 16 + l16;
#pragma unroll
      for (int r = 0; r < 8; ++r)
        Gbuf[(r + 8 * hi) * GSTR + colb] = acc[r];
    }
    __syncthreads();
    // --- selection phase: stream 32 candidates per thread into sorted top-20
    for (int ci = 0; ci < 32; ++ci) {
      const int col = ci * 16 + sub;
      const int j = ch * 512 + col;
      const float s = Gbuf[row * GSTR + col] - 0.5f * XXb[j];
      if (s > bv[KNN - 1]) {
        int p = KNN - 1;
        while (p > 0 && bv[p - 1] < s) {
          bv[p] = bv[p - 1]; bi[p] = bi[p - 1]; --p;
        }
        bv[p] = s; bi[p] = j;
      }
    }
    __syncthreads();
  }

  // --- merge: dump 16 sorted lists per row, then one thread per row k-way merges
  {
    const int base = (row * 16 + sub) * KNN;
    for (int m = 0; m < KNN; ++m) { Mv[base + m] = bv[m]; Mi[base + m] = bi[m]; }
  }
  __syncthreads();
  if (tid < 16) {
    int p[16];
    for (int s2 = 0; s2 < 16; ++s2) p[s2] = 0;
    const size_t obase = ((size_t)b * NN + i0 + tid) * KNN;
    for (int m = 0; m < KNN; ++m) {
      float best = -3.4e38f; int bs = 0;
      for (int s2 = 0; s2 < 16; ++s2) {
        if (p[s2] < KNN) {
          float cv = Mv[(tid * 16 + s2) * KNN + p[s2]];
          if (cv > best) { best = cv; bs = s2; }
        }
      }
      IDX[obase + m] = Mi[(tid * 16 + bs) * KNN + p[bs]];
      ++p[bs];
    }
  }
}

// ---------------------------------------------------------------- K4: gather, zmax/zmin, BN partials
// 256 threads = 4 point-groups x 64 channels; o contiguous => coalesced 256B gathers of v.
__global__ void k_gather(const float* __restrict__ U, const float* __restrict__ V,
                         const int* __restrict__ IDX,
                         float* __restrict__ ZMAX, float* __restrict__ ZMIN,
                         float* __restrict__ STAT) {
  __shared__ float rs[256], rq[256];
  const int tid = threadIdx.x;
  const int g = tid >> 6, o = tid & 63;
  const int pos = blockIdx.x * 4 + g;     // pos = b*N + n
  const int b = pos >> 12;
  const float uval = U[(size_t)pos * 64 + o];
  float smax = -3.4e38f, smin = 3.4e38f, sum = 0.0f, sq = 0.0f;
  const int* ip = IDX + (size_t)pos * KNN;
#pragma unroll
  for (int m = 0; m < KNN; ++m) {
    const int j = ip[m];
    const float val = V[((size_t)b * NN + j) * 64 + o];
    smax = fmaxf(smax, val);
    smin = fminf(smin, val);
    const float y = val + uval;  // BN statistics are over pre-activation y
    sum += y; sq += y * y;
  }
  ZMAX[(size_t)pos * 64 + o] = smax + uval;
  ZMIN[(size_t)pos * 64 + o] = smin + uval;
  rs[tid] = sum; rq[tid] = sq;
  __syncthreads();
  if (tid < 64) {
    const float ts = rs[tid] + rs[64 + tid] + rs[128 + tid] + rs[192 + tid];
    const float tq = rq[tid] + rq[64 + tid] + rq[128 + tid] + rq[192 + tid];
    atomicAdd(&STAT[tid], ts);
    atomicAdd(&STAT[64 + tid], tq);
  }
}

// ---------------------------------------------------------------- K5: BN scale/bias per channel
__global__ void k_scale(const float* __restrict__ STAT, const float* __restrict__ gamma,
                        const float* __restrict__ beta, float* __restrict__ SCB) {
  const int o = threadIdx.x;
  const float inv = 1.0f / (float)((size_t)BB * NN * KNN);
  const float mean = STAT[o] * inv;
  const float var  = STAT[64 + o] * inv - mean * mean;
  const float sc   = gamma[o] * rsqrtf(var + BN_EPSF);
  SCB[o]      = sc;
  SCB[64 + o] = beta[o] - mean * sc;
}

// ---------------------------------------------------------------- K6: affine + leaky + max-over-k
// Monotone activation => max over k is act(affine(.)) evaluated at zmax or zmin.
__global__ void k_out(const float* __restrict__ ZMAX, const float* __restrict__ ZMIN,
                      const float* __restrict__ SCB, float* __restrict__ out) {
  const int n = blockIdx.x * 256 + threadIdx.x;
  const int o = blockIdx.y, b = blockIdx.z;
  const float sc = SCB[o], bs = SCB[64 + o];
  const size_t pos = (size_t)b * NN + n;
  float y1 = sc * ZMAX[pos * 64 + o] + bs; y1 = (y1 >= 0.0f) ? y1 : SLOPEF * y1;
  float y2 = sc * ZMIN[pos * 64 + o] + bs; y2 = (y2 >= 0.0f) ? y2 : SLOPEF * y2;
  out[((size_t)b * OO + o) * NN + n] = fmaxf(y1, y2);
}

// ---------------------------------------------------------------- launch
extern "C" void kernel_launch(void* const* d_in, const int* in_sizes, int n_in,
                              void* d_out, int out_size, void* d_ws, size_t ws_size,
                              hipStream_t stream) {
  (void)in_sizes; (void)n_in; (void)out_size; (void)ws_size;
  const float* X     = (const float*)d_in[0];  // (8,64,4096)
  const float* W     = (const float*)d_in[1];  // (64,128)
  const float* gamma = (const float*)d_in[2];  // (64)
  const float* beta  = (const float*)d_in[3];  // (64)
  float* out = (float*)d_out;                  // (8,64,4096)

  // Workspace layout (floats); total ~36.3 MB.
  float* ws   = (float*)d_ws;
  float* XX   = ws;                            // B*N
  int*   IDX  = (int*)(ws + BB * NN);          // B*N*KNN
  float* U    = ws + BB * NN + BB * NN * KNN;  // B*N*O
  float* V    = U + (size_t)BB * NN * OO;      // B*N*O
  float* ZMAX = V + (size_t)BB * NN * OO;      // B*N*O
  float* ZMIN = ZMAX + (size_t)BB * NN * OO;   // B*N*O
  float* STAT = ZMIN + (size_t)BB * NN * OO;   // 128
  float* SCB  = STAT + 128;                    // 128

  k_zero  <<<dim3(1),               dim3(128), 0, stream>>>(STAT);
  k_xx    <<<dim3((BB * NN) / 256), dim3(256), 0, stream>>>(X, XX);
  k_proj  <<<dim3(BB * 64),         dim3(128), 0, stream>>>(X, W, U, V);
  k_knn   <<<dim3(NN / 16, BB),     dim3(256), 0, stream>>>(X, XX, IDX);
  k_gather<<<dim3((BB * NN) / 4),   dim3(256), 0, stream>>>(U, V, IDX, ZMAX, ZMIN, STAT);
  k_scale <<<dim3(1),               dim3(64),  0, stream>>>(STAT, gamma, beta, SCB);
  k_out   <<<dim3(NN / 256, OO, BB), dim3(256), 0, stream>>>(ZMAX, ZMIN, SCB, out);
}